// RNNSequenceEncoder_84636625535256
// MI455X (gfx1250) — compile-verified
//
#include <hip/hip_runtime.h>
#include <hip/hip_bf16.h>

// RNNSequenceEncoder for MI455X (gfx1250): persistent-scan f32 WMMA kernel with
// async global->LDS double-buffered staging. BS=256, T=128, IN=128, STATE=1024, OUT=1024.

#define BS_    256
#define T_     128
#define IN_    128
#define STATE_ 1024
#define OUT_   1024

#define KCHUNK       128          // K-chunk of A staged in LDS per phase
#define CHUNKS       (STATE_ / KCHUNK)      // 8
#define AT_STRIDE    40           // 32 rows + 8 pad -> conflict-free ds fragment reads
#define LDS_X_STRIDE 132          // 128 + 4 pad
#define BUF_FLOATS   (KCHUNK * AT_STRIDE)   // 5120 floats = 20 KB per buffer

#define GRID_X 8                  // state/out column groups of 128
#define GRID_Y 8                  // batch groups of 32
#define NBLOCKS (GRID_X * GRID_Y)

typedef __attribute__((ext_vector_type(2))) float v2f;
typedef __attribute__((ext_vector_type(8))) float v8f;

// LDS byte address of a __shared__-derived pointer: per ISA, the LDS aperture
// maps to the workgroup's LDS space via the low 32 address bits.
__device__ __forceinline__ unsigned lds_addr_u32(const void* p)
{
    return (unsigned)(uintptr_t)p;
}

// One element of an async (transposing) global->LDS copy; tracked by ASYNCcnt.
__device__ __forceinline__ void async_ld_lds_b32(unsigned lds_addr, const void* gaddr)
{
    asm volatile("global_load_async_to_lds_b32 %0, %1, off"
                 :: "v"(lds_addr), "v"(gaddr)
                 : "memory");
}

__device__ __forceinline__ void wait_async_and_barrier()
{
    asm volatile("s_wait_asynccnt 0x0" ::: "memory");
    __syncthreads();
}

// ---------------- software grid barrier (all 64 blocks resident) ----------------
__device__ __forceinline__ void grid_barrier(unsigned* cnt, unsigned* gen, unsigned nblocks)
{
    __syncthreads();
    if (threadIdx.x == 0) {
        __threadfence();   // make this block's global writes device-visible
        unsigned g = __hip_atomic_load(gen, __ATOMIC_RELAXED, __HIP_MEMORY_SCOPE_AGENT);
        unsigned arrived =
            __hip_atomic_fetch_add(cnt, 1u, __ATOMIC_ACQ_REL, __HIP_MEMORY_SCOPE_AGENT);
        if (arrived == nblocks - 1u) {
            __hip_atomic_store(cnt, 0u, __ATOMIC_RELAXED, __HIP_MEMORY_SCOPE_AGENT);
            __hip_atomic_fetch_add(gen, 1u, __ATOMIC_ACQ_REL, __HIP_MEMORY_SCOPE_AGENT);
        } else {
            while (__hip_atomic_load(gen, __ATOMIC_ACQUIRE, __HIP_MEMORY_SCOPE_AGENT) == g) {
                __builtin_amdgcn_s_sleep(2);
            }
        }
    }
    __syncthreads();
}

// Issue async transposed staging of one K-chunk of src (row-major 256x1024)
// into LDS buf laid out [k_local][m] with stride AT_STRIDE.
__device__ __forceinline__ void stage_chunk_async(const float* __restrict__ src,
                                                  float* buf, int batchBase, int c, int tid)
{
    for (int i = tid; i < 32 * KCHUNK; i += 256) {
        const int m = i >> 7;                 // / KCHUNK (=128)
        const int k = i & (KCHUNK - 1);
        async_ld_lds_b32(lds_addr_u32(&buf[k * AT_STRIDE + m]),
                         src + (size_t)(batchBase + m) * STATE_ + (size_t)c * KCHUNK + k);
    }
}

// One K-chunk of the main GEMM: acc{0,1} += A_chunk(2 m-tiles) x Wrow-fragment.
__device__ __forceinline__ void gemm_chunk(const float* __restrict__ buf,
                                           const float* __restrict__ wrow,   // &W[n*STATE + c*KCHUNK]
                                           int l16, int half, v8f& acc0, v8f& acc1)
{
    for (int k0 = 0; k0 < KCHUNK; k0 += 4) {
        const int ks = k0 + half * 2;
        v2f a0, a1, b;
        b.x = wrow[ks];
        b.y = wrow[ks + 1];
        a0.x = buf[ks * AT_STRIDE + l16];
        a0.y = buf[(ks + 1) * AT_STRIDE + l16];
        a1.x = buf[ks * AT_STRIDE + 16 + l16];
        a1.y = buf[(ks + 1) * AT_STRIDE + 16 + l16];
        acc0 = __builtin_amdgcn_wmma_f32_16x16x4_f32(false, a0, false, b, (short)0, acc0, false, false);
        acc1 = __builtin_amdgcn_wmma_f32_16x16x4_f32(false, a1, false, b, (short)0, acc1, false, false);
    }
}

// Compute this wave's ext tile for timestep t (two 16-row tiles of
// x[batch..][t][:].W_inT columns s0..s0+16). Stages x via async copy into ldsX.
__device__ __forceinline__ void ext_tile(const float* __restrict__ x,
                                         const float* __restrict__ W_in,
                                         float* ldsX, int t, int batchBase, int s0,
                                         int l16, int half, int tid,
                                         v8f& out0, v8f& out1)
{
    __syncthreads();   // previous LDS users finished
    for (int i = tid; i < 32 * IN_; i += 256) {
        const int m = i >> 7;
        const int k = i & (IN_ - 1);
        async_ld_lds_b32(lds_addr_u32(&ldsX[m * LDS_X_STRIDE + k]),
                         x + ((size_t)(batchBase + m) * T_ + (size_t)t) * IN_ + k);
    }
    wait_async_and_barrier();

    v8f e0 = {}, e1 = {};
    for (int k0 = 0; k0 < IN_; k0 += 4) {
        const int ks = k0 + half * 2;
        v2f a0, a1, b;
        const float* wp = W_in + (size_t)(s0 + l16) * IN_ + ks;
        b.x = wp[0];
        b.y = wp[1];
        a0.x = ldsX[l16 * LDS_X_STRIDE + ks];
        a0.y = ldsX[l16 * LDS_X_STRIDE + ks + 1];
        a1.x = ldsX[(16 + l16) * LDS_X_STRIDE + ks];
        a1.y = ldsX[(16 + l16) * LDS_X_STRIDE + ks + 1];
        e0 = __builtin_amdgcn_wmma_f32_16x16x4_f32(false, a0, false, b, (short)0, e0, false, false);
        e1 = __builtin_amdgcn_wmma_f32_16x16x4_f32(false, a1, false, b, (short)0, e1, false, false);
    }
    out0 = e0;
    out1 = e1;
}

// Persistent scan:  A_0 = ext_0 + b_in;  for t: S = ReLU(A.W_recT + b_rec);
// A <- S + ext_{t+1} + b_in (fused epilogue); then readout P = S.W_outT + b_out.
// grid = (8 col groups of 128, 8 batch groups of 32), block = 256 (8 waves).
// A chunks are staged into double-buffered LDS via async copies overlapping WMMA.
__global__ __launch_bounds__(256)
void rnn_scan_kernel(const float* __restrict__ x,
                     const float* __restrict__ W_in,  const float* __restrict__ b_in,
                     const float* __restrict__ W_rec, const float* __restrict__ b_rec,
                     const float* __restrict__ W_out, const float* __restrict__ b_out,
                     float* __restrict__ A_ws,   // 256x1024 combined input (per step)
                     float* __restrict__ S_ws,   // 256x1024 final state
                     float* __restrict__ P_ws,   // 256x1024 readout
                     unsigned* barCnt, unsigned* barGen)
{
    __shared__ float lds[2 * BUF_FLOATS];     // two 20 KB A^T buffers; x tile aliases buf0
    float* buf0 = lds;
    float* buf1 = lds + BUF_FLOATS;
    float* ldsX = lds;

    const int tid  = threadIdx.x;
    const int lane = tid & 31;
    const int half = lane >> 4;
    const int l16  = lane & 15;
    const int wave = tid >> 5;

    const int batchBase = blockIdx.y * 32;
    const int colBase   = blockIdx.x * 128;
    const int n0        = colBase + wave * 16;   // this wave's column tile
    const int n         = n0 + l16;

    // ---- A_0 = ext_0 + b_in ----
    {
        v8f e0, e1;
        ext_tile(x, W_in, ldsX, 0, batchBase, n0, l16, half, tid, e0, e1);
        const float bin = b_in[n];
        #pragma unroll
        for (int v = 0; v < 8; ++v) {
            const int m = v + half * 8;
            A_ws[(size_t)(batchBase + m) * STATE_ + n]      = e0[v] + bin;
            A_ws[(size_t)(batchBase + 16 + m) * STATE_ + n] = e1[v] + bin;
        }
    }
    grid_barrier(barCnt, barGen, NBLOCKS);

    // ---- recurrent scan: one grid barrier per timestep ----
    for (int t = 0; t < T_; ++t) {
        v8f acc0 = {}, acc1 = {};
        // preload chunk 0
        stage_chunk_async(A_ws, buf0, batchBase, 0, tid);
        wait_async_and_barrier();
        for (int c = 0; c < CHUNKS; ++c) {
            float* cur = (c & 1) ? buf1 : buf0;
            float* nxt = (c & 1) ? buf0 : buf1;
            if (c + 1 < CHUNKS)
                stage_chunk_async(A_ws, nxt, batchBase, c + 1, tid);  // overlap with WMMA
            __builtin_prefetch(W_rec + (size_t)n * STATE_ + (size_t)(c + 1) * KCHUNK, 0, 1);
            gemm_chunk(cur, W_rec + (size_t)n * STATE_ + (size_t)c * KCHUNK,
                       l16, half, acc0, acc1);
            wait_async_and_barrier();   // next chunk landed; all waves done with cur
        }

        // S_{t+1} tile = ReLU(acc + b_rec)
        const float brec = b_rec[n];
        float s0v[8], s1v[8];
        #pragma unroll
        for (int v = 0; v < 8; ++v) {
            float u0 = acc0[v] + brec;
            float u1 = acc1[v] + brec;
            s0v[v] = u0 > 0.0f ? u0 : 0.0f;
            s1v[v] = u1 > 0.0f ? u1 : 0.0f;
        }

        if (t < T_ - 1) {
            // fused epilogue: A_{t+1} = S_{t+1} + ext_{t+1} + b_in (this block's tile only)
            v8f e0, e1;
            ext_tile(x, W_in, ldsX, t + 1, batchBase, n0, l16, half, tid, e0, e1);
            const float bin = b_in[n];
            #pragma unroll
            for (int v = 0; v < 8; ++v) {
                const int m = v + half * 8;
                A_ws[(size_t)(batchBase + m) * STATE_ + n]      = s0v[v] + e0[v] + bin;
                A_ws[(size_t)(batchBase + 16 + m) * STATE_ + n] = s1v[v] + e1[v] + bin;
            }
        } else {
            #pragma unroll
            for (int v = 0; v < 8; ++v) {
                const int m = v + half * 8;
                S_ws[(size_t)(batchBase + m) * STATE_ + n]      = s0v[v];
                S_ws[(size_t)(batchBase + 16 + m) * STATE_ + n] = s1v[v];
            }
        }
        grid_barrier(barCnt, barGen, NBLOCKS);
    }

    // ---- readout: P = S . W_outT + b_out (same double-buffered pipeline) ----
    {
        v8f acc0 = {}, acc1 = {};
        stage_chunk_async(S_ws, buf0, batchBase, 0, tid);
        wait_async_and_barrier();
        for (int c = 0; c < CHUNKS; ++c) {
            float* cur = (c & 1) ? buf1 : buf0;
            float* nxt = (c & 1) ? buf0 : buf1;
            if (c + 1 < CHUNKS)
                stage_chunk_async(S_ws, nxt, batchBase, c + 1, tid);
            gemm_chunk(cur, W_out + (size_t)n * STATE_ + (size_t)c * KCHUNK,
                       l16, half, acc0, acc1);
            wait_async_and_barrier();
        }
        const float bo = b_out[n];
        #pragma unroll
        for (int v = 0; v < 8; ++v) {
            const int m = v + half * 8;
            P_ws[(size_t)(batchBase + m) * OUT_ + n]      = acc0[v] + bo;
            P_ws[(size_t)(batchBase + 16 + m) * OUT_ + n] = acc1[v] + bo;
        }
    }
}

__global__ void rnn_init_barrier_kernel(unsigned* bar)
{
    bar[0] = 0u;   // count
    bar[1] = 0u;   // generation
}

// Broadcast P[b][o] across T: out[b][t][o] = P[b][o]; 134 MB coalesced float4
// stores -> HBM-write bound (~5.8 us at 23.3 TB/s).
__global__ __launch_bounds__(256)
void rnn_broadcast_kernel(const float4* __restrict__ P, float4* __restrict__ out)
{
    const size_t total = (size_t)BS_ * T_ * (OUT_ / 4);     // 8,388,608 float4
    for (size_t i = (size_t)blockIdx.x * blockDim.x + threadIdx.x;
         i < total; i += (size_t)gridDim.x * blockDim.x) {
        const size_t o4 = i & (OUT_ / 4 - 1);
        const size_t b  = i >> 15;                          // / (T_ * OUT_/4)
        out[i] = P[b * (OUT_ / 4) + o4];
    }
}

extern "C" void kernel_launch(void* const* d_in, const int* in_sizes, int n_in,
                              void* d_out, int out_size, void* d_ws, size_t ws_size,
                              hipStream_t stream) {
    const float* x     = (const float*)d_in[0];
    const float* W_in  = (const float*)d_in[1];
    const float* b_in  = (const float*)d_in[2];
    const float* W_rec = (const float*)d_in[3];
    const float* b_rec = (const float*)d_in[4];
    const float* W_out = (const float*)d_in[5];
    const float* b_out = (const float*)d_in[6];
    float* out = (float*)d_out;

    // workspace: A (1MB) + S (1MB) + P (1MB) + barrier counters
    float* A_ws = (float*)d_ws;
    float* S_ws = A_ws + (size_t)BS_ * STATE_;
    float* P_ws = S_ws + (size_t)BS_ * STATE_;
    unsigned* bar = (unsigned*)(P_ws + (size_t)BS_ * OUT_);

    rnn_init_barrier_kernel<<<1, 1, 0, stream>>>(bar);
    rnn_scan_kernel<<<dim3(GRID_X, GRID_Y), 256, 0, stream>>>(
        x, W_in, b_in, W_rec, b_rec, W_out, b_out,
        A_ws, S_ws, P_ws, bar, bar + 1);
    rnn_broadcast_kernel<<<8192, 256, 0, stream>>>((const float4*)P_ws, (float4*)out);
}